// MambaBlock_72730976190811
// MI455X (gfx1250) — compile-verified
//
#include <hip/hip_runtime.h>
#include <math.h>

// ---------------- problem constants (from reference) ----------------
static constexpr int BATCH   = 2;
static constexpr int SEQ     = 1024;
static constexpr int DMODEL  = 768;
static constexpr int DINNER  = 1536;          // 2*DMODEL
static constexpr int XZW     = 2 * DINNER;    // 3072
static constexpr int NSTATE  = 16;
static constexpr int DCONV   = 4;
static constexpr int DTRANK  = 96;            // ceil(DINNER/16)
static constexpr int PROJW   = DTRANK + 2 * NSTATE;  // 128
static constexpr int MTOK    = BATCH * SEQ;   // 2048 tokens

typedef __attribute__((ext_vector_type(2))) float v2f;
typedef __attribute__((ext_vector_type(8))) float v8f;

__device__ __forceinline__ float siluf(float v)     { return v / (1.0f + __expf(-v)); }
__device__ __forceinline__ float softplusf(float v) { return (v > 20.0f) ? v : log1pf(__expf(v)); }

__device__ __forceinline__ v8f wmma_f32(v2f a, v2f b, v8f c) {
    return __builtin_amdgcn_wmma_f32_16x16x4_f32(
        /*neg_a=*/false, a, /*neg_b=*/false, b,
        /*c_mod=*/(short)0, c, /*reuse_a=*/false, /*reuse_b=*/false);
}

// ---------------- RMSNorm: one block per token ----------------
__global__ __launch_bounds__(256)
void rmsnorm_kernel(const float* __restrict__ x, const float* __restrict__ w,
                    float* __restrict__ xn)
{
    __shared__ float red[8];
    const int row = blockIdx.x;
    const float* xr = x + (size_t)row * DMODEL;
    float ss = 0.0f;
    for (int i = threadIdx.x; i < DMODEL; i += 256) { float v = xr[i]; ss += v * v; }
    #pragma unroll
    for (int off = 16; off; off >>= 1) ss += __shfl_xor(ss, off, 32);
    if ((threadIdx.x & 31) == 0) red[threadIdx.x >> 5] = ss;
    __syncthreads();
    if (threadIdx.x < 8) {
        float v = red[threadIdx.x];
        #pragma unroll
        for (int off = 4; off; off >>= 1) v += __shfl_xor(v, off, 8);
        if (threadIdx.x == 0) red[0] = v;
    }
    __syncthreads();
    const float scale = rsqrtf(red[0] * (1.0f / DMODEL) + 1e-5f);
    for (int i = threadIdx.x; i < DMODEL; i += 256)
        xn[(size_t)row * DMODEL + i] = xr[i] * scale * w[i];
}

// ---------------- Generic fp32 WMMA GEMM: C = act(A@B + bias) + resid ----------------
// Row-major A(M x K) lda, B(K x N) ldb, C(M x N) ldc.
// One wave computes a 16x64 strip (4 accumulators) via V_WMMA_F32_16X16X4_F32:
// the A fragment is loaded once per k-step and fed to 4 WMMAs; the 4 B fragments
// per row are 64 consecutive floats across 16 lanes (coalesced). A block of 8
// waves covers 128 rows x 64 cols, all sharing the same B panel (L0/L2 reuse).
// act: 0 = none, 1 = softplus. bias (len N) and resid (M x N, stride ldr) optional.
__global__ __launch_bounds__(256)
void gemm_wmma_f32(const float* __restrict__ A, int lda,
                   const float* __restrict__ Bm, int ldb,
                   float* __restrict__ C, int ldc,
                   int M, int N, int K,
                   const float* __restrict__ bias,
                   const float* __restrict__ resid, int ldr,
                   int act)
{
    const int lane  = threadIdx.x & 31;
    const int wave  = threadIdx.x >> 5;
    const int tileM = blockIdx.y * 128 + wave * 16;
    const int tileN = blockIdx.x * 64;
    if (tileM >= M) return;                     // wave-uniform scalar branch (EXEC stays full)

    const int half = lane >> 4;                 // 0: K pair {0,1}; 1: K pair {2,3}
    const int lidx = lane & 15;

    // A fragment: row = tileM+lidx, k = 4*kk + 2*half + {0,1}
    const float* aptr = A + (size_t)(tileM + lidx) * lda + 2 * half;
    // B fragments: cols tileN + 16*t + lidx, rows k = 4*kk + 2*half + {0,1}
    const float* bptr = Bm + (size_t)(2 * half) * ldb + (tileN + lidx);

    v8f acc0 = {}, acc1 = {}, acc2 = {}, acc3 = {};
    #pragma unroll 2
    for (int k = 0; k < K; k += 4) {
        const float2 av = *reinterpret_cast<const float2*>(aptr + k); // 8B aligned: lda even, k%4==0
        v2f a; a[0] = av.x; a[1] = av.y;

        const float* b0 = bptr + (size_t)k * ldb;        // row 2*half + k
        const float* b1 = b0 + ldb;                      // row 2*half + k + 1
        v2f f0, f1, f2, f3;
        f0[0] = b0[ 0]; f0[1] = b1[ 0];
        f1[0] = b0[16]; f1[1] = b1[16];
        f2[0] = b0[32]; f2[1] = b1[32];
        f3[0] = b0[48]; f3[1] = b1[48];

        acc0 = wmma_f32(a, f0, acc0);
        acc1 = wmma_f32(a, f1, acc1);
        acc2 = wmma_f32(a, f2, acc2);
        acc3 = wmma_f32(a, f3, acc3);
    }

    // C/D layout: element j -> row = tileM + j + 8*half, col = subtile base + lidx
    v8f accs[4] = {acc0, acc1, acc2, acc3};
    #pragma unroll
    for (int t = 0; t < 4; ++t) {
        const int col = tileN + 16 * t + lidx;
        #pragma unroll
        for (int j = 0; j < 8; ++j) {
            const int row = tileM + j + 8 * half;
            float v = accs[t][j];
            if (bias)     v += bias[col];
            if (act == 1) v  = softplusf(v);
            if (resid)    v += resid[(size_t)row * ldr + col];
            C[(size_t)row * ldc + col] = v;
        }
    }
}

// ---------------- Depthwise causal conv1d(width 4) + bias + SiLU ----------------
// x_in = columns [0, DINNER) of xz (row stride XZW).
__global__ __launch_bounds__(256)
void conv_silu_kernel(const float* __restrict__ xz, const float* __restrict__ cw,
                      const float* __restrict__ cb, float* __restrict__ u)
{
    const int idx = blockIdx.x * 256 + threadIdx.x;
    if (idx >= BATCH * SEQ * DINNER) return;
    const int d = idx % DINNER;
    const int l = (idx / DINNER) % SEQ;
    const int b = idx / (DINNER * SEQ);

    const float* base = xz + (size_t)b * SEQ * XZW + d;
    float acc = cb[d];
    #pragma unroll
    for (int j = 0; j < DCONV; ++j) {
        const int ll = l - (DCONV - 1) + j;
        if (ll >= 0) acc += cw[d * DCONV + j] * base[(size_t)ll * XZW];
    }
    u[idx] = siluf(acc);
}

// ---------------- Selective scan (reference log-cumsum formulation) ----------------
// One lane per (b, d, n); 16 consecutive lanes share (b, d). Wave32 -> two channels/wave.
// y[b,l,d] = sum_n C[b,l,n] * exp(min(logR,20)) * S  +  u*D
__global__ __launch_bounds__(256)
void scan_kernel(const float* __restrict__ delta, const float* __restrict__ u,
                 const float* __restrict__ proj,  const float* __restrict__ A_log,
                 const float* __restrict__ Dp,    float* __restrict__ y)
{
    const int t = blockIdx.x * 256 + threadIdx.x;   // < BATCH*DINNER*NSTATE = 49152
    const int n = t & (NSTATE - 1);
    const int d = (t >> 4) % DINNER;
    const int b = t / (NSTATE * DINNER);

    const float Acoef = -__expf(A_log[d * NSTATE + n]);
    const float dcoef = Dp[d];
    const float* drow = delta + (size_t)b * SEQ * DINNER + d;
    const float* urow = u     + (size_t)b * SEQ * DINNER + d;
    const float* prow = proj  + (size_t)b * SEQ * PROJW;
    float*       yrow = y     + (size_t)b * SEQ * DINNER + d;

    float logR = 0.0f, S = 0.0f;
    for (int l = 0; l < SEQ; ++l) {
        const float dv = drow[(size_t)l * DINNER];             // broadcast within 16-lane group
        const float uv = urow[(size_t)l * DINNER];
        const float Bv = prow[l * PROJW + DTRANK + n];
        const float Cv = prow[l * PROJW + DTRANK + NSTATE + n];

        // log_dA clipped to [-7, 20]; since floor -7 > log(1e-7), log(clip(dA,1e-7)) == log_dA
        const float logdA = fminf(fmaxf(dv * Acoef, -7.0f), 20.0f);
        logR += logdA;
        S += dv * Bv * uv * __expf(fminf(-logR, 20.0f));
        const float h = __expf(fminf(logR, 20.0f)) * S;

        float yp = Cv * h;
        #pragma unroll
        for (int off = 8; off; off >>= 1) yp += __shfl_xor(yp, off, 16);
        if (n == 0) yrow[(size_t)l * DINNER] = yp + uv * dcoef;
    }
}

// ---------------- Gate: y *= silu(z)   (z = cols [DINNER, 2*DINNER) of xz) ----------------
__global__ __launch_bounds__(256)
void gate_kernel(float* __restrict__ y, const float* __restrict__ xz)
{
    const int idx = blockIdx.x * 256 + threadIdx.x;
    if (idx >= MTOK * DINNER) return;
    const int m = idx / DINNER;
    const int d = idx % DINNER;
    const float zv = xz[(size_t)m * XZW + DINNER + d];
    y[idx] *= siluf(zv);
}

// ---------------- launch ----------------
extern "C" void kernel_launch(void* const* d_in, const int* in_sizes, int n_in,
                              void* d_out, int out_size, void* d_ws, size_t ws_size,
                              hipStream_t stream)
{
    const float* x       = (const float*)d_in[0];
    const float* w_norm  = (const float*)d_in[1];
    const float* w_in    = (const float*)d_in[2];
    const float* conv_w  = (const float*)d_in[3];
    const float* conv_b  = (const float*)d_in[4];
    const float* w_xproj = (const float*)d_in[5];
    const float* w_dt    = (const float*)d_in[6];
    const float* b_dt    = (const float*)d_in[7];
    const float* A_log   = (const float*)d_in[8];
    const float* D_param = (const float*)d_in[9];
    const float* w_out   = (const float*)d_in[10];
    float* out = (float*)d_out;

    // workspace layout (floats), every region fully written before read
    float* ws    = (float*)d_ws;
    float* xn    = ws;                                 // 2048*768
    float* xz    = xn    + (size_t)MTOK * DMODEL;      // 2048*3072
    float* u     = xz    + (size_t)MTOK * XZW;         // 2048*1536
    float* proj  = u     + (size_t)MTOK * DINNER;      // 2048*128
    float* delta = proj  + (size_t)MTOK * PROJW;       // 2048*1536
    float* y     = delta + (size_t)MTOK * DINNER;      // 2048*1536
    (void)in_sizes; (void)n_in; (void)out_size; (void)ws_size;

    // 1) RMSNorm
    rmsnorm_kernel<<<MTOK, 256, 0, stream>>>(x, w_norm, xn);

    // 2) xz = xn @ w_in        (2048x768 @ 768x3072)
    gemm_wmma_f32<<<dim3(XZW / 64, MTOK / 128), 256, 0, stream>>>(
        xn, DMODEL, w_in, XZW, xz, XZW, MTOK, XZW, DMODEL,
        nullptr, nullptr, 0, 0);

    // 3) u = silu(causal_conv(x_in) + conv_b)
    {
        const int total = BATCH * SEQ * DINNER;
        conv_silu_kernel<<<(total + 255) / 256, 256, 0, stream>>>(xz, conv_w, conv_b, u);
    }

    // 4) proj = u @ w_xproj    (2048x1536 @ 1536x128)
    gemm_wmma_f32<<<dim3(PROJW / 64, MTOK / 128), 256, 0, stream>>>(
        u, DINNER, w_xproj, PROJW, proj, PROJW, MTOK, PROJW, DINNER,
        nullptr, nullptr, 0, 0);

    // 5) delta = softplus(dt_i @ w_dt + b_dt)   (dt_i = proj[:, :96], lda = 128)
    gemm_wmma_f32<<<dim3(DINNER / 64, MTOK / 128), 256, 0, stream>>>(
        proj, PROJW, w_dt, DINNER, delta, DINNER, MTOK, DINNER, DTRANK,
        b_dt, nullptr, 0, 1);

    // 6) selective scan -> y (= y_ssm)
    {
        const int total = BATCH * DINNER * NSTATE;     // 49152
        scan_kernel<<<total / 256, 256, 0, stream>>>(delta, u, proj, A_log, D_param, y);
    }

    // 7) y *= silu(z)
    {
        const int total = MTOK * DINNER;
        gate_kernel<<<(total + 255) / 256, 256, 0, stream>>>(y, xz);
    }

    // 8) out = x + y @ w_out   (2048x1536 @ 1536x768, residual add)
    gemm_wmma_f32<<<dim3(DMODEL / 64, MTOK / 128), 256, 0, stream>>>(
        y, DINNER, w_out, DMODEL, out, DMODEL, MTOK, DMODEL, DINNER,
        nullptr, x, DMODEL, 0);
}